// GBottleneck_50165218017977
// MI455X (gfx1250) — compile-verified
//
#include <hip/hip_runtime.h>

typedef __attribute__((ext_vector_type(2))) float v2f;
typedef __attribute__((ext_vector_type(8))) float v8f;

// ---------------------------------------------------------------------------
// Weight pre-pack: P holds, per 64-column strip, per K-group g (4 rows), per
// half kh (lane 0-15 vs 16-31), per column c, the float2 fragment
//   { W[4g+kh][strip*64+c] , W[4g+2+kh][strip*64+c] }
// so the GEMM inner loop reads one 64-bit LDS word per B fragment.
// float index: (((strip*(K/4)+g)*2 + kh)*64 + c)*2 + sel
// ---------------------------------------------------------------------------
__global__ void pack_w(const float* __restrict__ W, float* __restrict__ P, int K)
{
    int idx = blockIdx.x * blockDim.x + threadIdx.x;   // over K*256
    if (idx >= K * 256) return;
    int k = idx >> 8, n = idx & 255;
    int g = k >> 2,  j = k & 3;
    int kh  = j & 1;          // which half-lane group
    int sel = j >> 1;         // .x or .y
    int strip = n >> 6, c = n & 63;
    P[((((strip * (K >> 2) + g) << 1) + kh) * 64 + c) * 2 + sel] = W[idx];
}

// ---------------------------------------------------------------------------
// WMMA f32 GEMM:  Y[M,256] = A[M,K] @ W[K,256] (+ bias) , W given packed (P).
// Block = 128 threads = 4 waves; wave -> 16(M) x 64(N) tile, 4 accumulators.
// B fragments served from LDS (one ds_load_b64 each); A via global b64.
// ---------------------------------------------------------------------------
template <int K>
__global__ __launch_bounds__(128) void wmma_gemm_f32(
    const float* __restrict__ A, const float* __restrict__ P,
    const float* __restrict__ bias, float* __restrict__ Y, int M)
{
    constexpr int N   = 256;
    constexpr int ENT = (K / 4) * 2 * 64;      // v2f entries per 64-col strip
    __shared__ v2f ldsB[ENT];

    // Cooperative stage of this block's packed strip (K*64 floats).
    {
        const float4* src = (const float4*)(P + (size_t)blockIdx.y * (K * 64));
        float4* dst = (float4*)ldsB;
        for (int i = threadIdx.x; i < ENT / 2; i += 128) dst[i] = src[i];
    }
    __syncthreads();

    const int lane = threadIdx.x & 31;
    const int wave = threadIdx.x >> 5;
    const int m0   = (blockIdx.x * 4 + wave) * 16;
    const int n0   = blockIdx.y * 64;
    const int lm   = lane & 15;
    const int kh   = lane >> 4;

    v8f acc0 = {}, acc1 = {}, acc2 = {}, acc3 = {};

    const float* arow = A + (size_t)(m0 + lm) * K + (kh << 1);
    const v2f*   brow = ldsB + kh * 64 + lm;

    for (int g = 0; g < K / 4; ++g) {
        v2f a = *(const v2f*)(arow + 4 * g);
        const v2f* bp = brow + g * 128;
        v2f b0 = bp[0];
        v2f b1 = bp[16];
        v2f b2 = bp[32];
        v2f b3 = bp[48];
        acc0 = __builtin_amdgcn_wmma_f32_16x16x4_f32(false, a, false, b0, (short)0, acc0, false, false);
        acc1 = __builtin_amdgcn_wmma_f32_16x16x4_f32(false, a, false, b1, (short)0, acc1, false, false);
        acc2 = __builtin_amdgcn_wmma_f32_16x16x4_f32(false, a, false, b2, (short)0, acc2, false, false);
        acc3 = __builtin_amdgcn_wmma_f32_16x16x4_f32(false, a, false, b3, (short)0, acc3, false, false);
    }

    float bv0 = 0.f, bv1 = 0.f, bv2 = 0.f, bv3 = 0.f;
    if (bias) {
        bv0 = bias[n0 + lm];
        bv1 = bias[n0 + 16 + lm];
        bv2 = bias[n0 + 32 + lm];
        bv3 = bias[n0 + 48 + lm];
    }

    // C/D layout: VGPR i -> row m0 + i + 8*(lane>=16), col n0 + (lane&15) (+16t)
    float* yb = Y + (size_t)(m0 + (kh << 3)) * N + n0 + lm;
#pragma unroll
    for (int i = 0; i < 8; ++i) {
        float* yr = yb + (size_t)i * N;
        yr[0]  = acc0[i] + bv0;
        yr[16] = acc1[i] + bv1;
        yr[32] = acc2[i] + bv2;
        yr[48] = acc3[i] + bv3;
    }
}

// ---------------------------------------------------------------------------
// Edge scatter: Y[b, dst, :] += ew[e] * S[b, src, :]   (DH = 256, B = 8)
// One thread per (edge, batch, 4-float chunk); f32 atomics land in L2
// (support + aggregation arrays are 82 MB each, L2-resident at 192 MB).
// ---------------------------------------------------------------------------
__global__ void edge_scatter(const float* __restrict__ S,
                             const int* __restrict__ esrc,
                             const int* __restrict__ edst,
                             const float* __restrict__ ew,
                             float* __restrict__ Y, int E, int Nn)
{
    const long per_edge = 8L * 64;   // B * (DH/4)
    long idx = (long)blockIdx.x * blockDim.x + threadIdx.x;
    if (idx >= (long)E * per_edge) return;
    int e  = (int)(idx / per_edge);
    int r  = (int)(idx % per_edge);
    int b  = r >> 6;
    int d4 = (r & 63) << 2;

    float w = ew[e];
    long srow = ((long)b * Nn + esrc[e]) * 256 + d4;
    long drow = ((long)b * Nn + edst[e]) * 256 + d4;
    float4 v = *(const float4*)(S + srow);
    atomicAdd(Y + drow + 0, v.x * w);
    atomicAdd(Y + drow + 1, v.y * w);
    atomicAdd(Y + drow + 2, v.z * w);
    atomicAdd(Y + drow + 3, v.w * w);
}

// h = (h + t) * 0.5
__global__ void resid_avg(float* __restrict__ h, const float* __restrict__ t, long n)
{
    long i = (long)blockIdx.x * blockDim.x + threadIdx.x;
    if (i < n) h[i] = (h[i] + t[i]) * 0.5f;
}

// Tiny GEMM for the DH=256 -> DOUT=3 layer: one thread per (row, j)
__global__ void small_gemm(const float* __restrict__ X, const float* __restrict__ W,
                           const float* __restrict__ bias, float* __restrict__ Y,
                           int M, int K)
{
    long idx = (long)blockIdx.x * blockDim.x + threadIdx.x;
    if (idx >= (long)M * 3) return;
    int m = (int)(idx / 3), j = (int)(idx % 3);
    float s = bias ? bias[j] : 0.f;
    const float* xr = X + (size_t)m * K;
    for (int k = 0; k < K; ++k) s += xr[k] * W[k * 3 + j];
    Y[idx] = s;
}

// Scatter for DOUT=3
__global__ void edge_scatter3(const float* __restrict__ S,
                              const int* __restrict__ esrc,
                              const int* __restrict__ edst,
                              const float* __restrict__ ew,
                              float* __restrict__ Y, int E, int Nn)
{
    long idx = (long)blockIdx.x * blockDim.x + threadIdx.x;
    if (idx >= (long)E * 24) return;
    int e = (int)(idx / 24);
    int r = (int)(idx % 24);
    int b = r / 3, j = r % 3;
    float w = ew[e];
    atomicAdd(Y + ((long)b * Nn + edst[e]) * 3 + j,
              w * S[((long)b * Nn + esrc[e]) * 3 + j]);
}

extern "C" void kernel_launch(void* const* d_in, const int* in_sizes, int n_in,
                              void* d_out, int out_size, void* d_ws, size_t ws_size,
                              hipStream_t stream)
{
    const float* x     = (const float*)d_in[0];
    const int*   esrc  = (const int*)d_in[1];
    const int*   edst  = (const int*)d_in[2];
    const float* ew    = (const float*)d_in[3];
    const float* W_in  = (const float*)d_in[4];
    const float* LW_in = (const float*)d_in[5];
    const float* b_in  = (const float*)d_in[6];
    const float* W_blk = (const float*)d_in[7];   // [6,2,256,256]
    const float* LW_blk= (const float*)d_in[8];
    const float* b_blk = (const float*)d_in[9];   // [6,2,256]
    const float* W_out = (const float*)d_in[10];  // [256,3]
    const float* LW_out= (const float*)d_in[11];
    const float* b_out = (const float*)d_in[12];

    const int B = 8, Nn = 10000, DIN = 512, DH = 256, E = 160000, L = 6;
    const int M = B * Nn;                 // 80000 rows
    const long HN = (long)M * DH;         // 20,480,000 elems

    float* xout = (float*)d_out;               // [M,3]
    float* h    = (float*)d_out + (size_t)M*3; // [M,256] lives in d_out
    float* S    = (float*)d_ws;                // support scratch
    float* T    = S  + HN;
    float* T2   = T  + HN;
    float* P    = T2 + HN;                     // packed weights (<= 512 KB)

    const dim3 gemmGrid(M / 64, DH / 64);
    const long scat_tot = (long)E * 8 * 64;
    const int  scat_blk = (int)((scat_tot + 255) / 256);

    auto gemm = [&](const float* xin, int K, const float* Wm, const float* bias, float* y) {
        pack_w<<<(K * 256 + 255) / 256, 256, 0, stream>>>(Wm, P, K);
        if (K == 512)
            wmma_gemm_f32<512><<<gemmGrid, 128, 0, stream>>>(xin, P, bias, y, M);
        else
            wmma_gemm_f32<256><<<gemmGrid, 128, 0, stream>>>(xin, P, bias, y, M);
    };

    auto gconv = [&](const float* xin, int K, const float* Wm, const float* LWm,
                     const float* bias, float* y) {
        gemm(xin, K, Wm, nullptr, S);
        gemm(xin, K, LWm, bias, y);
        edge_scatter<<<scat_blk, 256, 0, stream>>>(S, esrc, edst, ew, y, E, Nn);
    };

    // Input layer: x [M,512] -> h
    gconv(x, DIN, W_in, LW_in, b_in, h);

    // 6 residual blocks
    for (int l = 0; l < L; ++l) {
        const float* W0  = W_blk  + (size_t)(l*2 + 0) * DH * DH;
        const float* LW0 = LW_blk + (size_t)(l*2 + 0) * DH * DH;
        const float* bb0 = b_blk  + (size_t)(l*2 + 0) * DH;
        const float* W1  = W_blk  + (size_t)(l*2 + 1) * DH * DH;
        const float* LW1 = LW_blk + (size_t)(l*2 + 1) * DH * DH;
        const float* bb1 = b_blk  + (size_t)(l*2 + 1) * DH;
        gconv(h, DH, W0, LW0, bb0, T);
        gconv(T, DH, W1, LW1, bb1, T2);
        resid_avg<<<(int)((HN + 255) / 256), 256, 0, stream>>>(h, T2, HN);
    }

    // Output layer (DH -> 3): reuse S as [M,3] support scratch
    const long m3 = (long)M * 3;
    small_gemm<<<(int)((m3 + 255) / 256), 256, 0, stream>>>(h, W_out, nullptr, S, M, DH);
    small_gemm<<<(int)((m3 + 255) / 256), 256, 0, stream>>>(h, LW_out, b_out, xout, M, DH);
    const long s3 = (long)E * 24;
    edge_scatter3<<<(int)((s3 + 255) / 256), 256, 0, stream>>>(S, esrc, edst, ew, xout, E, Nn);
}